// HyperGNN_74139725463574
// MI455X (gfx1250) — compile-verified
//
#include <hip/hip_runtime.h>

#define N_NODES 20000
#define N_EDGES 640000
#define DIM     64
// matvec pipeline: 2 waves/block, 4 nodes/wave, double-buffered 16KB LDS tiles
#define MV_WAVES_PER_BLOCK 2
#define MV_BLOCKS          2500
#define MV_TOTAL_WAVES     (MV_BLOCKS * MV_WAVES_PER_BLOCK)   // 5000
#define MV_NODES_PER_WAVE  (N_NODES / MV_TOTAL_WAVES)          // 4
#define MV_LDS_BYTES       (MV_WAVES_PER_BLOCK * 2 * DIM * DIM * 4) // 64 KB

typedef __attribute__((ext_vector_type(2))) float v2f;
typedef __attribute__((ext_vector_type(8))) float v8f;
typedef __attribute__((ext_vector_type(4))) unsigned v4u;
typedef __attribute__((ext_vector_type(4))) int v4i;
typedef __attribute__((ext_vector_type(8))) int v8i;

// ---------- helpers: order-preserving float<->uint for atomicMax ----------
__device__ __forceinline__ unsigned orderKey(float f) {
    unsigned b = __float_as_uint(f);
    return (b & 0x80000000u) ? ~b : (b | 0x80000000u);
}
__device__ __forceinline__ float invKey(unsigned u) {
    return __uint_as_float((u & 0x80000000u) ? (u & 0x7FFFFFFFu) : ~u);
}

// ---------- TDM: DMA one 64x64 fp32 weight tile (16KB) global -> LDS ----------
// D# per CDNA5 ISA ch.8: group0 {count=1 | lds_addr | global_addr | type=2},
// group1 {data_size=4B, tensor 64x64, tile 64x64, strides 64 / 4096}.
__device__ __forceinline__ void tdm_load_w_tile(const float* wsrc, unsigned lds_byte_off) {
    unsigned long long ga = (unsigned long long)(uintptr_t)wsrc;
    v4u g0;
    g0[0] = 1u;                                          // count=1, user descriptor
    g0[1] = lds_byte_off;                                // lds_addr (bytes)
    g0[2] = (unsigned)(ga & 0xFFFFFFFFull);              // global_addr[31:0]
    g0[3] = (unsigned)((ga >> 32) & 0x01FFFFFFull)       // global_addr[56:32]
          | (2u << 30);                                  // type = 2 ("image")
    v8i g1;
    g1[0] = (int)(2u << 16);        // wg_mask=0, data_size=2 (4 bytes)
    g1[1] = (int)(64u << 16);       // tensor_dim0 = 64 (low 16 bits @ bit 48)
    g1[2] = (int)(64u << 16);       // tensor_dim1 = 64 (low 16 bits @ bit 80)
    g1[3] = (int)(64u << 16);       // tile_dim0   = 64 (@ bit 112)
    g1[4] = (int)64;                // tile_dim1   = 64 (@ bit 128); tile_dim2=0
    g1[5] = (int)64;                // tensor_dim0_stride = 64 (elements)
    g1[6] = (int)(4096u << 16);     // tensor_dim1_stride = 4096 (low 16 @ bit 208)
    g1[7] = 0;
    v4i g2 = {0, 0, 0, 0};
    v4i g3 = {0, 0, 0, 0};
#if defined(__clang_major__) && (__clang_major__ >= 23)
    v8i gx = {0, 0, 0, 0, 0, 0, 0, 0};
    __builtin_amdgcn_tensor_load_to_lds(g0, g1, g2, g3, gx, 0);
#else
    __builtin_amdgcn_tensor_load_to_lds(g0, g1, g2, g3, 0);
#endif
}

// ---------- one node's h[n] = act(x[n]) @ W[n] from an LDS-resident W tile ----
// K-step outer loop, 4 column-tile accumulators share one A operand per step.
template <bool DO_RELU>
__device__ __forceinline__ void node_matvec_from_lds(const float* __restrict__ xn,
                                                     const float* __restrict__ Wl,
                                                     float* __restrict__ hout,
                                                     int lane) {
    const int half = lane >> 4;                 // 0: K pair (r,r+1) | 1: (r+2,r+3)
    const int l    = lane & 15;
    const float amask = (l == 0) ? 1.0f : 0.0f; // only row M=0 of A is live
    v8f c0 = {}, c1 = {}, c2 = {}, c3 = {};
    #pragma unroll
    for (int s = 0; s < 16; ++s) {
        const int r = 4 * s + 2 * half;
        float xv0 = xn[r], xv1 = xn[r + 1];
        if (DO_RELU) { xv0 = fmaxf(xv0, 0.0f); xv1 = fmaxf(xv1, 0.0f); }
        v2f a; a.x = amask * xv0; a.y = amask * xv1;
        const float* r0 = Wl + r * DIM + l;     // LDS: ds_read2-mergeable pairs
        const float* r1 = r0 + DIM;
        v2f b;
        b.x = r0[0];  b.y = r1[0];
        c0 = __builtin_amdgcn_wmma_f32_16x16x4_f32(false, a, false, b, (short)0, c0, false, false);
        b.x = r0[16]; b.y = r1[16];
        c1 = __builtin_amdgcn_wmma_f32_16x16x4_f32(false, a, false, b, (short)0, c1, false, false);
        b.x = r0[32]; b.y = r1[32];
        c2 = __builtin_amdgcn_wmma_f32_16x16x4_f32(false, a, false, b, (short)0, c2, false, false);
        b.x = r0[48]; b.y = r1[48];
        c3 = __builtin_amdgcn_wmma_f32_16x16x4_f32(false, a, false, b, (short)0, c3, false, false);
    }
    if (half == 0) {                            // D row 0 lives in c[0], lanes 0..15
        hout[l]      = c0[0];
        hout[16 + l] = c1[0];
        hout[32 + l] = c2[0];
        hout[48 + l] = c3[0];
    }
}

// ---------- per-node matvec: TDM-prefetched, double-buffered, WMMA compute ----
template <bool DO_RELU>
__global__ __launch_bounds__(MV_WAVES_PER_BLOCK * 32)
void matvec_wmma_tdm_kernel(const float* __restrict__ x,
                            const float* __restrict__ w,
                            float* __restrict__ h) {
    extern __shared__ float smem[];
    const int lane   = threadIdx.x & 31;
    const int waveid = __builtin_amdgcn_readfirstlane((int)(threadIdx.x >> 5));
    const int gwave  = blockIdx.x * MV_WAVES_PER_BLOCK + waveid;   // [0, 5000)

    float* buf = smem + waveid * 2 * DIM * DIM;                    // two 16KB tiles
    const unsigned buf_off = (unsigned)(uintptr_t)buf;             // LDS byte offset

    // prologue: prefetch first node's W
    tdm_load_w_tile(w + (size_t)gwave * DIM * DIM, buf_off);

    int cur = 0;
    #pragma unroll 1
    for (int i = 0; i < MV_NODES_PER_WAVE; ++i) {
        const int node = gwave + i * MV_TOTAL_WAVES;
        if (i + 1 < MV_NODES_PER_WAVE) {
            const int nxt = node + MV_TOTAL_WAVES;
            tdm_load_w_tile(w + (size_t)nxt * DIM * DIM,
                            buf_off + (unsigned)((cur ^ 1) * DIM * DIM * 4));
            __builtin_amdgcn_s_wait_tensorcnt(1);   // wait current, keep next in flight
        } else {
            __builtin_amdgcn_s_wait_tensorcnt(0);
        }
        node_matvec_from_lds<DO_RELU>(x + (size_t)node * DIM,
                                      buf + cur * DIM * DIM,
                                      h + (size_t)node * DIM,
                                      lane);
        cur ^= 1;
    }
}

// ---------- init kernels ----------
__global__ void init_nodes_kernel(unsigned* __restrict__ nmax,
                                  float* __restrict__ nsum, int n) {
    int i = blockIdx.x * blockDim.x + threadIdx.x;
    if (i < n) { nmax[i] = 0u; nsum[i] = 0.0f; }
}
__global__ void copy_bias_kernel(const float* __restrict__ b,
                                 float* __restrict__ out, int n) {
    int i = blockIdx.x * blockDim.x + threadIdx.x;
    if (i < n) out[i] = b[i];
}

// ---------- per-edge score + leaky relu + segment max ----------
__global__ __launch_bounds__(256)
void edge_score_kernel(const float* __restrict__ h,
                       const int* __restrict__ row,
                       const int* __restrict__ col,
                       const float* __restrict__ a,
                       float* __restrict__ scores,
                       unsigned* __restrict__ nmax, int n_edges) {
    int e = blockIdx.x * blockDim.x + threadIdx.x;
    if (e >= n_edges) return;
    const int r = row[e], c = col[e];
    const float4* __restrict__ hi = (const float4*)(h + (size_t)r * DIM);
    const float4* __restrict__ hj = (const float4*)(h + (size_t)c * DIM);
    const float4* __restrict__ as = (const float4*)(a + (size_t)r * 2 * DIM);
    float s = 0.0f;
    #pragma unroll
    for (int q = 0; q < 16; ++q) {
        float4 u = hi[q], v = as[q];
        s += u.x * v.x + u.y * v.y + u.z * v.z + u.w * v.w;
    }
    #pragma unroll
    for (int q = 0; q < 16; ++q) {
        float4 u = hj[q], v = as[16 + q];
        s += u.x * v.x + u.y * v.y + u.z * v.z + u.w * v.w;
    }
    s = (s > 0.0f) ? s : 0.2f * s;             // leaky_relu, slope 0.2
    scores[e] = s;
    atomicMax(&nmax[c], orderKey(s));
}

// ---------- exp(score - segmax) + segment sum (in-place on scores) ----------
__global__ __launch_bounds__(256)
void edge_exp_kernel(float* __restrict__ scores,
                     const int* __restrict__ col,
                     const unsigned* __restrict__ nmax,
                     float* __restrict__ nsum, int n_edges) {
    int e = blockIdx.x * blockDim.x + threadIdx.x;
    if (e >= n_edges) return;
    const int c = col[e];
    float ex = __expf(scores[e] - invKey(nmax[c]));
    scores[e] = ex;
    atomicAdd(&nsum[c], ex);
}

// ---------- normalize + weighted scatter: out[col] += h[row] * att ----------
__global__ __launch_bounds__(256)
void edge_scatter_kernel(const float* __restrict__ h,
                         const float* __restrict__ ex,
                         const int* __restrict__ row,
                         const int* __restrict__ col,
                         const float* __restrict__ nsum,
                         float* __restrict__ out, int n_edges) {
    int tid = blockIdx.x * blockDim.x + threadIdx.x;
    int e = tid >> 4;
    if (e >= n_edges) return;
    const int q = tid & 15;
    const int r = row[e], c = col[e];
    const float att = ex[e] / nsum[c];
    float4 hv = ((const float4*)(h + (size_t)r * DIM))[q];
    float* o = out + (size_t)c * DIM + q * 4;
    atomicAdd(o + 0, hv.x * att);
    atomicAdd(o + 1, hv.y * att);
    atomicAdd(o + 2, hv.z * att);
    atomicAdd(o + 3, hv.w * att);
}

// ---------- host-side layer driver ----------
static void run_gat_layer(const float* xin, const float* w, const float* a,
                          const float* b, float* h_buf, float* scores,
                          unsigned* nmax, float* nsum, float* out,
                          const int* row, const int* col, int do_relu,
                          hipStream_t stream) {
    const int N = N_NODES, E = N_EDGES;
    if (do_relu)
        matvec_wmma_tdm_kernel<true><<<MV_BLOCKS, MV_WAVES_PER_BLOCK * 32,
                                       MV_LDS_BYTES, stream>>>(xin, w, h_buf);
    else
        matvec_wmma_tdm_kernel<false><<<MV_BLOCKS, MV_WAVES_PER_BLOCK * 32,
                                        MV_LDS_BYTES, stream>>>(xin, w, h_buf);
    init_nodes_kernel<<<(N + 255) / 256, 256, 0, stream>>>(nmax, nsum, N);
    copy_bias_kernel<<<(N * DIM + 255) / 256, 256, 0, stream>>>(b, out, N * DIM);
    edge_score_kernel<<<(E + 255) / 256, 256, 0, stream>>>(h_buf, row, col, a,
                                                           scores, nmax, E);
    edge_exp_kernel<<<(E + 255) / 256, 256, 0, stream>>>(scores, col, nmax, nsum, E);
    edge_scatter_kernel<<<(E * 16 + 255) / 256, 256, 0, stream>>>(
        h_buf, scores, row, col, nsum, out, E);
}

extern "C" void kernel_launch(void* const* d_in, const int* in_sizes, int n_in,
                              void* d_out, int out_size, void* d_ws, size_t ws_size,
                              hipStream_t stream) {
    (void)in_sizes; (void)n_in; (void)out_size; (void)ws_size;
    const float* x  = (const float*)d_in[0];
    const int*   ei = (const int*)d_in[1];      // (2, E): row then col
    const float* w0 = (const float*)d_in[2];
    const float* a0 = (const float*)d_in[3];
    const float* b0 = (const float*)d_in[4];
    const float* w1 = (const float*)d_in[5];
    const float* a1 = (const float*)d_in[6];
    const float* b1 = (const float*)d_in[7];
    const int* row = ei;
    const int* col = ei + N_EDGES;

    // workspace carve-up (all 16B-aligned given 256B-aligned d_ws)
    float*    h_buf   = (float*)d_ws;                       // N*64
    float*    tmp_out = h_buf + (size_t)N_NODES * DIM;      // N*64 (layer-1 out)
    float*    scores  = tmp_out + (size_t)N_NODES * DIM;    // E
    unsigned* nmax    = (unsigned*)(scores + N_EDGES);      // N
    float*    nsum    = (float*)(nmax + N_NODES);           // N

    // layer 1: h0 = x @ W0 ; out0 = GAT(h0) + b0   (stored pre-relu)
    run_gat_layer(x, w0, a0, b0, h_buf, scores, nmax, nsum, tmp_out,
                  row, col, /*do_relu=*/0, stream);
    // layer 2: h1 = relu(out0) @ W1 ; out = GAT(h1) + b1
    run_gat_layer(tmp_out, w1, a1, b1, h_buf, scores, nmax, nsum, (float*)d_out,
                  row, col, /*do_relu=*/1, stream);
}